// GNNLayer_35347580846828
// MI455X (gfx1250) — compile-verified
//
#include <hip/hip_runtime.h>
#include <hip/hip_bf16.h>

typedef float v2f __attribute__((ext_vector_type(2)));
typedef float v8f __attribute__((ext_vector_type(8)));

#define D_DIM 64
#define LN_EPS 1e-5f

// ---------------------------------------------------------------------------
// 0) init: acc = 0 over N*64, deg = 1.0 over N (self-loop contributes +1)
// ---------------------------------------------------------------------------
__global__ void init_ws_kernel(float* __restrict__ acc, float* __restrict__ deg, int N) {
    int tid = blockIdx.x * blockDim.x + threadIdx.x;
    int total = N * D_DIM;
    if (tid < total) acc[tid] = 0.0f;
    if (tid < N)     deg[tid] = 1.0f;
}

// ---------------------------------------------------------------------------
// 1) degree accumulation over real edges (dst row of edge_index)
// ---------------------------------------------------------------------------
__global__ void deg_kernel(const int* __restrict__ dst, float* __restrict__ deg, int E) {
    int e = blockIdx.x * blockDim.x + threadIdx.x;
    if (e < E) atomicAdd(&deg[dst[e]], 1.0f);
}

// ---------------------------------------------------------------------------
// 2) dinv = rsqrt(deg)   (deg >= 1 always, due to self loop)
// ---------------------------------------------------------------------------
__global__ void dinv_kernel(float* __restrict__ deg, int N) {
    int i = blockIdx.x * blockDim.x + threadIdx.x;
    if (i < N) deg[i] = rsqrtf(deg[i]);
}

// ---------------------------------------------------------------------------
// 3) xw = x @ W  via v_wmma_f32_16x16x4_f32
//    Block = 128 threads = 4 waves. Block handles 16 rows; wave w handles
//    output columns [16w, 16w+16). K loop: 16 chained WMMAs with K=4.
//    ISA layouts (32-bit, wave32):
//      A 16x4 : lane = h*16 + m  (h=lane>>4), VGPR0=K(2h), VGPR1=K(2h+1)
//      B 4x16 : lane = h*16 + n,              VGPR0=K(2h), VGPR1=K(2h+1)
//      D 16x16: VGPR r -> row (8h + r), col n
// ---------------------------------------------------------------------------
__global__ __launch_bounds__(128) void gemm_wmma_kernel(const float* __restrict__ x,
                                                        const float* __restrict__ W,
                                                        float* __restrict__ xw) {
    const int lane  = threadIdx.x & 31;
    const int wave  = threadIdx.x >> 5;      // n-tile 0..3
    const int h     = lane >> 4;             // 0 or 1
    const int l16   = lane & 15;
    const int mbase = blockIdx.x * 16;
    const int m     = mbase + l16;           // A row for this lane
    const int n     = wave * 16 + l16;       // B/D column for this lane

    const float* __restrict__ xrow = x + (size_t)m * D_DIM;

    v8f c = {0.f, 0.f, 0.f, 0.f, 0.f, 0.f, 0.f, 0.f};
    #pragma unroll
    for (int kb = 0; kb < D_DIM; kb += 4) {
        const int k0 = kb + 2 * h;
        v2f a, b;
        a.x = xrow[k0];
        a.y = xrow[k0 + 1];
        b.x = W[(size_t)k0 * D_DIM + n];
        b.y = W[(size_t)(k0 + 1) * D_DIM + n];
        c = __builtin_amdgcn_wmma_f32_16x16x4_f32(
                /*neg_a=*/false, a, /*neg_b=*/false, b,
                /*c_mod=*/(short)0, c, /*reuse_a=*/false, /*reuse_b=*/false);
    }

    float* __restrict__ orow = xw + (size_t)(mbase + 8 * h) * D_DIM + n;
    #pragma unroll
    for (int r = 0; r < 8; ++r) {
        orow[(size_t)r * D_DIM] = c[r];
    }
}

// scalar tail for rows not covered by full 16-row WMMA tiles (EXEC must be all
// ones for WMMA, so remainder rows are handled here; N=50000 => empty grid)
__global__ void gemm_tail_kernel(const float* __restrict__ x,
                                 const float* __restrict__ W,
                                 float* __restrict__ xw, int row0, int N) {
    int t   = blockIdx.x * blockDim.x + threadIdx.x;
    int row = row0 + (t >> 6);
    int col = t & 63;
    if (row >= N) return;
    float s = 0.f;
    for (int k = 0; k < D_DIM; ++k)
        s += x[(size_t)row * D_DIM + k] * W[(size_t)k * D_DIM + col];
    xw[(size_t)row * D_DIM + col] = s;
}

// ---------------------------------------------------------------------------
// 4) edge scatter: acc[dst] += dinv[src]*dinv[dst] * xw[src]
//    ONE WAVE32 PER EDGE: per-edge index/dinv loads are wave-uniform (single
//    broadcast cacheline request); each lane handles 2 dims via a float2
//    (global_load_b64) gather, then 2 fp32 L2 atomics.
// ---------------------------------------------------------------------------
__global__ __launch_bounds__(256) void edge_scatter_kernel(const int* __restrict__ src,
                                                           const int* __restrict__ dst,
                                                           const float* __restrict__ dinv,
                                                           const float* __restrict__ xw,
                                                           float* __restrict__ acc, int E) {
    const int lane = threadIdx.x & 31;
    const int e    = blockIdx.x * 8 + (threadIdx.x >> 5);   // 8 waves per block
    if (e >= E) return;

    const int s = src[e];          // wave-uniform -> broadcast from one line
    const int q = dst[e];
    const float coeff = dinv[s] * dinv[q];

    const size_t gsrc = (size_t)s * D_DIM + lane * 2;
    const size_t gdst = (size_t)q * D_DIM + lane * 2;
    float2 m = *(const float2*)(xw + gsrc);                 // global_load_b64

    atomicAdd(&acc[gdst],     coeff * m.x);
    atomicAdd(&acc[gdst + 1], coeff * m.y);
}

// ---------------------------------------------------------------------------
// 5) finalize: z = acc + dinv^2*xw + b (self loop), h = x + z,
//    LayerNorm(h)*gamma+beta, ReLU. One wave32 per node, 2 dims/lane (float2).
// ---------------------------------------------------------------------------
__global__ __launch_bounds__(256) void finalize_kernel(const float* __restrict__ x,
                                                       const float* __restrict__ acc,
                                                       const float* __restrict__ xw,
                                                       const float* __restrict__ dinv,
                                                       const float* __restrict__ bias,
                                                       const float* __restrict__ gamma,
                                                       const float* __restrict__ beta,
                                                       float* __restrict__ out, int N) {
    const int lane = threadIdx.x & 31;
    const int node = blockIdx.x * 8 + (threadIdx.x >> 5);
    if (node >= N) return;

    const size_t base = (size_t)node * D_DIM + lane * 2;
    const float  di   = dinv[node];
    const float  di2  = di * di;

    float2 xv = *(const float2*)(x   + base);
    float2 av = *(const float2*)(acc + base);
    float2 wv = *(const float2*)(xw  + base);
    float2 bv = *(const float2*)(bias + lane * 2);

    float h0 = xv.x + av.x + di2 * wv.x + bv.x;
    float h1 = xv.y + av.y + di2 * wv.y + bv.y;

    // mean over 64 values (butterfly across wave32)
    float s = h0 + h1;
    #pragma unroll
    for (int o = 16; o > 0; o >>= 1) s += __shfl_xor(s, o, 32);
    const float mu = s * (1.0f / 64.0f);

    const float e0 = h0 - mu;
    const float e1 = h1 - mu;
    float v = e0 * e0 + e1 * e1;
    #pragma unroll
    for (int o = 16; o > 0; o >>= 1) v += __shfl_xor(v, o, 32);
    const float inv = rsqrtf(v * (1.0f / 64.0f) + LN_EPS);

    float2 g  = *(const float2*)(gamma + lane * 2);
    float2 be = *(const float2*)(beta  + lane * 2);

    float2 r;
    r.x = fmaxf(e0 * inv * g.x + be.x, 0.0f);
    r.y = fmaxf(e1 * inv * g.y + be.y, 0.0f);
    *(float2*)(out + base) = r;
}

// ---------------------------------------------------------------------------
extern "C" void kernel_launch(void* const* d_in, const int* in_sizes, int n_in,
                              void* d_out, int out_size, void* d_ws, size_t ws_size,
                              hipStream_t stream) {
    const float* x     = (const float*)d_in[0];
    const int*   ei    = (const int*)  d_in[1];   // [2, E]: row0 = src, row1 = dst
    const float* W     = (const float*)d_in[2];
    const float* bias  = (const float*)d_in[3];
    const float* gamma = (const float*)d_in[4];
    const float* beta  = (const float*)d_in[5];
    float*       out   = (float*)d_out;

    const int N = in_sizes[0] / D_DIM;
    const int E = in_sizes[1] / 2;

    const int* srcIdx = ei;
    const int* dstIdx = ei + E;

    // workspace layout (floats): xw [N*64] | acc [N*64] | deg/dinv [N]
    float* xw  = (float*)d_ws;
    float* acc = xw + (size_t)N * D_DIM;
    float* deg = acc + (size_t)N * D_DIM;

    // 0) init acc=0, deg=1
    {
        int total = N * D_DIM;
        init_ws_kernel<<<(total + 255) / 256, 256, 0, stream>>>(acc, deg, N);
    }
    // 1) degree
    deg_kernel<<<(E + 255) / 256, 256, 0, stream>>>(dstIdx, deg, E);
    // 2) dinv
    dinv_kernel<<<(N + 255) / 256, 256, 0, stream>>>(deg, N);
    // 3) xw = x @ W   (WMMA main + scalar tail)
    {
        int full_tiles = N / 16;
        if (full_tiles > 0)
            gemm_wmma_kernel<<<full_tiles, 128, 0, stream>>>(x, W, xw);
        int row0 = full_tiles * 16;
        int rem  = N - row0;
        if (rem > 0)
            gemm_tail_kernel<<<(rem * D_DIM + 255) / 256, 256, 0, stream>>>(x, W, xw, row0, N);
    }
    // 4) edge scatter: one wave32 per edge, float2 gathers, fp32 atomics
    edge_scatter_kernel<<<(E + 7) / 8, 256, 0, stream>>>(srcIdx, dstIdx, deg, xw, acc, E);
    // 5) fused self-loop + skip + LayerNorm + ReLU
    finalize_kernel<<<(N + 7) / 8, 256, 0, stream>>>(x, acc, xw, deg, bias, gamma, beta, out, N);
}